// SpectralDecomposition_19481971654664
// MI455X (gfx1250) — compile-verified
//
#include <hip/hip_runtime.h>

// ---------------------------------------------------------------------------
// Haar 2x2 decomposition of channel 0 of a (32,3,512,512) f32 tensor.
// Output: (32,4,256,256) f32  [ll, lh, hl, hh].
// Pure streaming problem: 33.5MB in + 33.5MB out => ~2.9us at 23.3 TB/s.
// Strategy: gfx1250 async global->LDS staging (B128 per lane), then packed
// VALU butterfly from LDS, non-temporal 128-bit stores.
// ---------------------------------------------------------------------------

typedef __attribute__((ext_vector_type(4))) float v4f;
typedef __attribute__((ext_vector_type(4))) int   v4i;

#define BLOCK 256

#if defined(__gfx1250__) && \
    __has_builtin(__builtin_amdgcn_global_load_async_to_lds_b128) && \
    __has_builtin(__builtin_amdgcn_s_wait_asynccnt)
#define HAAR_ASYNC 1
#else
#define HAAR_ASYNC 0
#endif

__device__ __forceinline__ void haar_butterfly(const v4f t0, const v4f t1,
                                               const v4f b0, const v4f b1,
                                               v4f& ll, v4f& lh, v4f& hl, v4f& hh) {
    // a = even cols of top row, b = odd cols of top row,
    // c = even cols of bottom row, d = odd cols of bottom row
    v4f A = { t0.x, t0.z, t1.x, t1.z };
    v4f B = { t0.y, t0.w, t1.y, t1.w };
    v4f C = { b0.x, b0.z, b1.x, b1.z };
    v4f D = { b0.y, b0.w, b1.y, b1.w };
    v4f s_ab = A + B;   // a+b
    v4f d_ab = A - B;   // a-b
    v4f s_cd = C + D;   // c+d
    v4f d_cd = C - D;   // c-d
    ll = (s_ab + s_cd) * 0.5f;   // (a+b+c+d)/2
    lh = (s_ab - s_cd) * 0.5f;   // (a+b-c-d)/2
    hl = (d_ab + d_cd) * 0.5f;   // (a-b+c-d)/2
    hh = (d_ab - d_cd) * 0.5f;   // (a-b-c+d)/2
}

__global__ __launch_bounds__(BLOCK) void haar2x2_kernel(const float* __restrict__ x,
                                                        float* __restrict__ out) {
    constexpr int W  = 512;   // input width
    constexpr int H  = 512;   // input height
    constexpr int C  = 3;     // input channels (we read only channel 0)
    constexpr int OW = 256;   // output width/height
    constexpr int SUB = OW * OW;          // 65536 elements per sub-band

    const int blk = blockIdx.x;           // 0..2047
    const int n   = blk >> 6;             // image index (0..31)
    const int rg  = blk & 63;             // row-group: 8 input rows / 4 output rows
    const int t   = threadIdx.x;          // 0..255

    // 8 input rows of channel 0 are contiguous: one 16KB slab.
    const float* src = x + (size_t)n * C * H * W + (size_t)rg * 8 * W;

    const int r = t >> 6;                 // local output row 0..3
    const int c = t & 63;                 // column group: 8 input cols / 4 output cols

    v4f t0, t1, b0, b1;

#if HAAR_ASYNC
    __shared__ alignas(16) float tile[8 * W];   // 16 KB
    {
        // Stage the whole 16KB slab: 1024 x 16B chunks, 4 per thread.
        auto gsrc = (__attribute__((address_space(1))) char*)src;
        auto ldst = (__attribute__((address_space(3))) char*)tile;
#pragma unroll
        for (int k = 0; k < 4; ++k) {
            const int ch = t + k * BLOCK;       // chunk id 0..1023
            __builtin_amdgcn_global_load_async_to_lds_b128(
                (__attribute__((address_space(1))) v4i*)(gsrc + (size_t)ch * 16),
                (__attribute__((address_space(3))) v4i*)(ldst + ch * 16),
                /*offset=*/0, /*cpol=*/0);
        }
    }
    __builtin_amdgcn_s_wait_asynccnt(0);
    __syncthreads();

    const float* top = &tile[(2 * r) * W + 8 * c];
    t0 = *(const v4f*)(top);
    t1 = *(const v4f*)(top + 4);
    b0 = *(const v4f*)(top + W);
    b1 = *(const v4f*)(top + W + 4);
#else
    // Fallback: direct non-temporal 128-bit loads (still fully coalesced).
    const float* top = src + (size_t)(2 * r) * W + 8 * c;
    t0 = __builtin_nontemporal_load((const v4f*)(top));
    t1 = __builtin_nontemporal_load((const v4f*)(top + 4));
    b0 = __builtin_nontemporal_load((const v4f*)(top + W));
    b1 = __builtin_nontemporal_load((const v4f*)(top + W + 4));
#endif

    v4f ll, lh, hl, hh;
    haar_butterfly(t0, t1, b0, b1, ll, lh, hl, hh);

    const int i = rg * 4 + r;             // output row 0..255
    float* o = out + (size_t)n * 4 * SUB + (size_t)i * OW + 4 * c;
    __builtin_nontemporal_store(ll, (v4f*)(o + 0 * SUB));
    __builtin_nontemporal_store(lh, (v4f*)(o + 1 * SUB));
    __builtin_nontemporal_store(hl, (v4f*)(o + 2 * SUB));
    __builtin_nontemporal_store(hh, (v4f*)(o + 3 * SUB));
}

extern "C" void kernel_launch(void* const* d_in, const int* in_sizes, int n_in,
                              void* d_out, int out_size, void* d_ws, size_t ws_size,
                              hipStream_t stream) {
    (void)in_sizes; (void)n_in; (void)out_size; (void)d_ws; (void)ws_size;
    const float* x = (const float*)d_in[0];
    float* out = (float*)d_out;
    // 32 images * 64 row-groups = 2048 blocks of 256 threads (8 wave32 each).
    haar2x2_kernel<<<dim3(2048), dim3(256), 0, stream>>>(x, out);
}